// GCN_55061480735304
// MI455X (gfx1250) — compile-verified
//
#include <hip/hip_runtime.h>

typedef __attribute__((ext_vector_type(2))) float v2f;
typedef __attribute__((ext_vector_type(8))) float v8f;

#define N_NODES 100000
#define IN_DIM  128
#define EMB     64
#define NCLS    18
#define NCLS_PAD 32

// ---------------------------------------------------------------------------
// Init: deg=1 (self-loop), agg1=0, logits region = b2 broadcast
// ---------------------------------------------------------------------------
__global__ void gcn_init(float* __restrict__ deg, float* __restrict__ agg1,
                         float* __restrict__ logits, const float* __restrict__ b2) {
  long long i = (long long)blockIdx.x * blockDim.x + threadIdx.x;
  if (i < (long long)N_NODES * EMB) agg1[i] = 0.0f;
  if (i < N_NODES) deg[i] = 1.0f;                       // self-loop contribution
  if (i < (long long)N_NODES * NCLS) logits[i] = b2[i % NCLS];
}

// ---------------------------------------------------------------------------
// Degree accumulation over edges (dst side)
// ---------------------------------------------------------------------------
__global__ void gcn_degree(const int* __restrict__ ei, int E, float* __restrict__ deg) {
  int e = blockIdx.x * blockDim.x + threadIdx.x;
  if (e < E) atomicAdd(&deg[ei[E + e]], 1.0f);
}

// deg -> dis = rsqrt(deg), in place (deg >= 1 always due to self-loops)
__global__ void gcn_rsqrt(float* __restrict__ d) {
  int i = blockIdx.x * blockDim.x + threadIdx.x;
  if (i < N_NODES) d[i] = rsqrtf(d[i]);
}

// ---------------------------------------------------------------------------
// GEMM1: h1[100000x64] = x[100000x128] * W1[128x64] via V_WMMA_F32_16X16X4_F32
// One wave per 16x16 output tile. 6250 M-tiles x 4 N-tiles = 25000 tiles.
// W1 staged in LDS (32KB) per block.
// ---------------------------------------------------------------------------
__global__ __launch_bounds__(256) void gcn_gemm1(const float* __restrict__ x,
                                                 const float* __restrict__ W1,
                                                 float* __restrict__ h1) {
  __shared__ float sW[IN_DIM * EMB];                    // 32 KB
  for (int i = threadIdx.x; i < IN_DIM * EMB; i += 256) sW[i] = W1[i];
  __syncthreads();

  const int lane = threadIdx.x & 31;
  const int wave = threadIdx.x >> 5;
  const int tile = blockIdx.x * 8 + wave;               // < 25000 exactly
  const int mt = tile >> 2;
  const int nt = tile & 3;

  const int mrow  = (mt << 4) + (lane & 15);            // A: M = lane&15
  const int ncol  = (nt << 4) + (lane & 15);            // B: N = lane&15
  const int khalf = (lane >> 4) << 1;                   // lanes 16-31 hold K+2,K+3

  const float* __restrict__ arow = x + (size_t)mrow * IN_DIM;

  v8f acc = {};
  for (int k0 = 0; k0 < IN_DIM; k0 += 4) {
    const int kk = k0 + khalf;
    v2f a = *(const v2f*)(arow + kk);                   // A[m][kk], A[m][kk+1]
    v2f b = { sW[kk * EMB + ncol], sW[(kk + 1) * EMB + ncol] };
    acc = __builtin_amdgcn_wmma_f32_16x16x4_f32(
        /*neg_a=*/false, a, /*neg_b=*/false, b,
        /*c_mod=*/(short)0, acc, /*reuse_a=*/false, /*reuse_b=*/false);
  }

  // C/D layout: VGPR r -> M = r (lanes 0-15) or M = r+8 (lanes 16-31)
  const int mbase = (mt << 4) + ((lane >> 4) << 3);
#pragma unroll
  for (int r = 0; r < 8; ++r)
    h1[(size_t)(mbase + r) * EMB + ncol] = acc[r];
}

// ---------------------------------------------------------------------------
// Scatter layer 1: one wave per edge, lane j handles columns 2j, 2j+1
// ---------------------------------------------------------------------------
__global__ __launch_bounds__(256) void gcn_scatter1(const int* __restrict__ ei, int E,
                                                    const float* __restrict__ dis,
                                                    const float* __restrict__ h1,
                                                    float* __restrict__ agg1) {
  const int lane = threadIdx.x & 31;
  const int e = blockIdx.x * 8 + (threadIdx.x >> 5);
  if (e >= E) return;
  const int s = ei[e];
  const int d = ei[E + e];
  const float w = dis[s] * dis[d];
  v2f m = *(const v2f*)(h1 + (size_t)s * EMB + lane * 2);
  float* o = agg1 + (size_t)d * EMB + lane * 2;
  atomicAdd(o,     m.x * w);
  atomicAdd(o + 1, m.y * w);
}

// ---------------------------------------------------------------------------
// Self-loop + bias + ReLU, writes r1 over agg1 in place
// ---------------------------------------------------------------------------
__global__ void gcn_relu1(const float* __restrict__ h1, const float* __restrict__ dis,
                          const float* __restrict__ b1, float* __restrict__ agg1) {
  long long i = (long long)blockIdx.x * blockDim.x + threadIdx.x;
  if (i >= (long long)N_NODES * EMB) return;
  const int node = (int)(i >> 6);
  const int col  = (int)(i & 63);
  float inv = dis[node]; inv *= inv;                    // self-loop norm = 1/deg
  float v = agg1[i] + h1[i] * inv + b1[col];
  agg1[i] = fmaxf(v, 0.0f);
}

// ---------------------------------------------------------------------------
// GEMM2: h2[100000x18] = r1[100000x64] * W2[64x18]; N padded to 32 in LDS.
// 6250 M-tiles x 2 N-tiles = 12500 tiles.
// ---------------------------------------------------------------------------
__global__ __launch_bounds__(256) void gcn_gemm2(const float* __restrict__ r1,
                                                 const float* __restrict__ W2,
                                                 float* __restrict__ h2) {
  __shared__ float sW[EMB * NCLS_PAD];                  // 8 KB, zero-padded
  for (int i = threadIdx.x; i < EMB * NCLS_PAD; i += 256) {
    const int r = i >> 5, c = i & 31;
    sW[i] = (c < NCLS) ? W2[r * NCLS + c] : 0.0f;
  }
  __syncthreads();

  const int lane = threadIdx.x & 31;
  const int tile = blockIdx.x * 8 + (threadIdx.x >> 5);
  if (tile < 6250 * 2) {
    const int mt = tile >> 1;
    const int nt = tile & 1;

    const int mrow  = (mt << 4) + (lane & 15);
    const int nloc  = (nt << 4) + (lane & 15);          // 0..31 (padded N)
    const int khalf = (lane >> 4) << 1;

    const float* __restrict__ arow = r1 + (size_t)mrow * EMB;

    v8f acc = {};
    for (int k0 = 0; k0 < EMB; k0 += 4) {
      const int kk = k0 + khalf;
      v2f a = *(const v2f*)(arow + kk);
      v2f b = { sW[kk * NCLS_PAD + nloc], sW[(kk + 1) * NCLS_PAD + nloc] };
      acc = __builtin_amdgcn_wmma_f32_16x16x4_f32(
          false, a, false, b, (short)0, acc, false, false);
    }

    if (nloc < NCLS) {
      const int mbase = (mt << 4) + ((lane >> 4) << 3);
#pragma unroll
      for (int r = 0; r < 8; ++r)
        h2[(size_t)(mbase + r) * NCLS + nloc] = acc[r];
    }
  }
}

// ---------------------------------------------------------------------------
// Scatter layer 2: one wave per edge, lanes 0..17 handle the 18 classes
// ---------------------------------------------------------------------------
__global__ __launch_bounds__(256) void gcn_scatter2(const int* __restrict__ ei, int E,
                                                    const float* __restrict__ dis,
                                                    const float* __restrict__ h2,
                                                    float* __restrict__ logits) {
  const int lane = threadIdx.x & 31;
  const int e = blockIdx.x * 8 + (threadIdx.x >> 5);
  if (e >= E) return;
  const int s = ei[e];
  const int d = ei[E + e];
  const float w = dis[s] * dis[d];
  if (lane < NCLS)
    atomicAdd(&logits[(size_t)d * NCLS + lane], h2[(size_t)s * NCLS + lane] * w);
}

// ---------------------------------------------------------------------------
// Self-loop for layer 2 + log_softmax. out[0:N*18) = log_softmax, logits follow.
// ---------------------------------------------------------------------------
__global__ void gcn_finalize(const float* __restrict__ h2, const float* __restrict__ dis,
                             float* __restrict__ out) {
  int i = blockIdx.x * blockDim.x + threadIdx.x;
  if (i >= N_NODES) return;
  float inv = dis[i]; inv *= inv;
  float* lg = out + (size_t)N_NODES * NCLS + (size_t)i * NCLS;   // partial logits (+b2)
  const float* hrow = h2 + (size_t)i * NCLS;
  float vals[NCLS];
  float mx = -INFINITY;
#pragma unroll
  for (int j = 0; j < NCLS; ++j) {
    float v = lg[j] + hrow[j] * inv;
    vals[j] = v;
    mx = fmaxf(mx, v);
  }
  float sum = 0.0f;
#pragma unroll
  for (int j = 0; j < NCLS; ++j) sum += expf(vals[j] - mx);
  const float lse = mx + logf(sum);
  float* ls = out + (size_t)i * NCLS;
#pragma unroll
  for (int j = 0; j < NCLS; ++j) {
    lg[j] = vals[j];                 // final logits
    ls[j] = vals[j] - lse;           // log_softmax
  }
}

// ---------------------------------------------------------------------------
extern "C" void kernel_launch(void* const* d_in, const int* in_sizes, int n_in,
                              void* d_out, int out_size, void* d_ws, size_t ws_size,
                              hipStream_t stream) {
  const float* x  = (const float*)d_in[0];
  const int*   ei = (const int*)d_in[1];    // edge_index [2, E] flattened
  const float* W1 = (const float*)d_in[2];
  const float* b1 = (const float*)d_in[3];
  const float* W2 = (const float*)d_in[4];
  const float* b2 = (const float*)d_in[5];
  float* out = (float*)d_out;

  const int E = in_sizes[1] / 2;            // 3,200,000

  // Workspace layout (floats): dis | h1 | agg1(r1) | h2   (~59 MB total)
  float* ws   = (float*)d_ws;
  float* dis  = ws;
  float* h1   = dis + N_NODES;
  float* agg1 = h1 + (size_t)N_NODES * EMB;
  float* h2   = agg1 + (size_t)N_NODES * EMB;
  float* logits = out + (size_t)N_NODES * NCLS;

  const int T = 256;

  gcn_init<<<(N_NODES * EMB + T - 1) / T, T, 0, stream>>>(dis, agg1, logits, b2);
  gcn_degree<<<(E + T - 1) / T, T, 0, stream>>>(ei, E, dis);
  gcn_rsqrt<<<(N_NODES + T - 1) / T, T, 0, stream>>>(dis);

  gcn_gemm1<<<25000 / 8, T, 0, stream>>>(x, W1, h1);                 // 3125 blocks
  gcn_scatter1<<<(E + 7) / 8, T, 0, stream>>>(ei, E, dis, h1, agg1);
  gcn_relu1<<<(N_NODES * EMB + T - 1) / T, T, 0, stream>>>(h1, dis, b1, agg1);

  gcn_gemm2<<<(12500 + 7) / 8, T, 0, stream>>>(agg1, W2, h2);
  gcn_scatter2<<<(E + 7) / 8, T, 0, stream>>>(ei, E, dis, h2, logits);
  gcn_finalize<<<(N_NODES + T - 1) / T, T, 0, stream>>>(h2, dis, out);
}